// TransformerEncoder_23210003268236
// MI455X (gfx1250) — compile-verified
//
#include <hip/hip_runtime.h>

#define D    128
#define NH   8
#define HDIM 16
#define QKV3 384
#define FFSZ 512

typedef float v2f __attribute__((ext_vector_type(2)));
typedef float v8f __attribute__((ext_vector_type(8)));

// ---------------------------------------------------------------------------
// C (Nrows x Ncols) = A (Nrows x K) * W^T  + bias   (W is Ncols x K, F.linear)
// One wave per 16x16 output tile; fp32 WMMA 16x16x4, K stepped by 4.
// A-frag (16x4 f32, 2 VGPR):  a[j] = A[lane&15][k0 + j + 2*(lane>>4)]
// B-frag (4x16 f32, 2 VGPR):  b[j] = B[k0 + j + 2*(lane>>4)][lane&15] = W[col][k]
// C/D  (16x16 f32, 8 VGPR):   c[i] -> row = i + 8*(lane>>4), col = lane&15
// ---------------------------------------------------------------------------
__global__ void gemm_bias_wmma(const float* __restrict__ A,
                               const float* __restrict__ W,
                               const float* __restrict__ bias,
                               float* __restrict__ C,
                               int K, int Ncols, int relu)
{
    const int lane = threadIdx.x & 31;
    const int wave = threadIdx.x >> 5;
    const int tm   = blockIdx.x;               // row tile
    const int tn   = blockIdx.y * 4 + wave;    // col tile
    const int row0 = tm << 4;
    const int col0 = tn << 4;
    const int r    = lane & 15;
    const int koff = (lane >> 4) << 1;         // 0 or 2

    const float* arow = A + (size_t)(row0 + r) * K + koff;
    const float* wrow = W + (size_t)(col0 + r) * K + koff;

    v8f c = {0.f, 0.f, 0.f, 0.f, 0.f, 0.f, 0.f, 0.f};
#pragma unroll 4
    for (int k0 = 0; k0 < K; k0 += 4) {
        v2f a = *(const v2f*)(arow + k0);
        v2f b = *(const v2f*)(wrow + k0);
        // (neg_a, A, neg_b, B, c_mod, C, reuse_a, reuse_b)
        c = __builtin_amdgcn_wmma_f32_16x16x4_f32(false, a, false, b,
                                                  (short)0, c, false, false);
    }

    const int   half = lane >> 4;
    const float bv   = bias[col0 + r];
#pragma unroll
    for (int i = 0; i < 8; ++i) {
        float val = c[i] + bv;
        if (relu) val = fmaxf(val, 0.f);
        C[(size_t)(row0 + i + (half << 3)) * Ncols + col0 + r] = val;
    }
}

// ---------------------------------------------------------------------------
// init: smax (encoded-uint) = 0 (== -inf under encoding), ssum = 0, agg = 0
// ---------------------------------------------------------------------------
__global__ void init_kernel(unsigned* __restrict__ smaxU,
                            float* __restrict__ ssum,
                            float* __restrict__ agg,
                            int n_head, int n_agg)
{
    int i = blockIdx.x * blockDim.x + threadIdx.x;
    if (i < n_head) { smaxU[i] = 0u; ssum[i] = 0.f; }
    if (i < n_agg)  { agg[i] = 0.f; }
}

__device__ __forceinline__ unsigned enc_f32(float x) {
    unsigned b = __float_as_uint(x);
    return (b & 0x80000000u) ? ~b : (b | 0x80000000u);
}
__device__ __forceinline__ float dec_f32(unsigned e) {
    unsigned b = (e & 0x80000000u) ? (e ^ 0x80000000u) : ~e;
    return __uint_as_float(b);
}

// ---------------------------------------------------------------------------
// per (edge, head): score = 0.25 * <q[tgt,h,:], k[src,h,:]>; atomicMax into smax
// qkv layout: row*384 : [0,128)=q [128,256)=k [256,384)=v
// ---------------------------------------------------------------------------
__global__ void edge_scores_kernel(const float* __restrict__ qkv,
                                   const int* __restrict__ edges,
                                   unsigned* __restrict__ smaxU,
                                   float* __restrict__ scores,
                                   int M)
{
    long tid = (long)blockIdx.x * blockDim.x + threadIdx.x;
    if (tid >= (long)M * NH) return;
    int e = (int)(tid >> 3), h = (int)(tid & 7);
    int s = edges[e];
    int t = edges[M + e];
    const float* q = qkv + (size_t)t * QKV3 + h * HDIM;
    const float* k = qkv + (size_t)s * QKV3 + D + h * HDIM;
    float acc = 0.f;
#pragma unroll
    for (int j = 0; j < 4; ++j) {
        float4 a = *(const float4*)(q + 4 * j);
        float4 b = *(const float4*)(k + 4 * j);
        acc += a.x * b.x + a.y * b.y + a.z * b.z + a.w * b.w;
    }
    float sc = acc * 0.25f;                 // HD^-0.5 = 1/4
    scores[(size_t)e * NH + h] = sc;
    atomicMax(smaxU + (size_t)t * NH + h, enc_f32(sc));
}

// ---------------------------------------------------------------------------
// per (edge, head): ex = exp(score - smax[tgt]); atomicAdd into ssum[tgt]
// ---------------------------------------------------------------------------
__global__ void edge_exp_kernel(const int* __restrict__ edges,
                                const unsigned* __restrict__ smaxU,
                                float* __restrict__ ssum,
                                float* __restrict__ ex,   // in: scores, out: exp
                                int M)
{
    long tid = (long)blockIdx.x * blockDim.x + threadIdx.x;
    if (tid >= (long)M * NH) return;
    int e = (int)(tid >> 3), h = (int)(tid & 7);
    int t = edges[M + e];
    float m  = dec_f32(smaxU[(size_t)t * NH + h]);
    float ev = expf(ex[(size_t)e * NH + h] - m);
    ex[(size_t)e * NH + h] = ev;
    atomicAdd(ssum + (size_t)t * NH + h, ev);
}

// ---------------------------------------------------------------------------
// per (edge, 4-dim chunk): agg[tgt] += v[src] * ex/(ssum[tgt]+eps)
// ---------------------------------------------------------------------------
__global__ void edge_agg_kernel(const float* __restrict__ qkv,
                                const int* __restrict__ edges,
                                const float* __restrict__ ex,
                                const float* __restrict__ ssum,
                                float* __restrict__ agg,
                                int M)
{
    long tid = (long)blockIdx.x * blockDim.x + threadIdx.x;
    if (tid >= (long)M * 32) return;
    int e = (int)(tid >> 5), c = (int)(tid & 31);
    int h = c >> 2;                              // 4 chunks per head
    int s = edges[e];
    int t = edges[M + e];
    float w = ex[(size_t)e * NH + h] /
              (ssum[(size_t)t * NH + h] + 1e-16f);
    float4 v = *(const float4*)(qkv + (size_t)s * QKV3 + 2 * D + c * 4);
    float* dst = agg + (size_t)t * D + c * 4;
    atomicAdd(dst + 0, v.x * w);
    atomicAdd(dst + 1, v.y * w);
    atomicAdd(dst + 2, v.z * w);
    atomicAdd(dst + 3, v.w * w);
}

// ---------------------------------------------------------------------------
// out[row] = LayerNorm(A[row] + B[row]; g, beta)   — one wave32 per 128-d row
// ---------------------------------------------------------------------------
__global__ void ln_residual_kernel(const float* __restrict__ A,
                                   const float* __restrict__ B,
                                   const float* __restrict__ g,
                                   const float* __restrict__ beta,
                                   float* __restrict__ out,
                                   int N)
{
    int row  = blockIdx.x * (blockDim.x >> 5) + (threadIdx.x >> 5);
    if (row >= N) return;
    int lane = threadIdx.x & 31;
    float4 a = *(const float4*)(A + (size_t)row * D + lane * 4);
    float4 b = *(const float4*)(B + (size_t)row * D + lane * 4);
    float t0 = a.x + b.x, t1 = a.y + b.y, t2 = a.z + b.z, t3 = a.w + b.w;
    float s  = t0 + t1 + t2 + t3;
    float ss = t0 * t0 + t1 * t1 + t2 * t2 + t3 * t3;
#pragma unroll
    for (int off = 16; off >= 1; off >>= 1) {
        s  += __shfl_xor(s,  off, 32);
        ss += __shfl_xor(ss, off, 32);
    }
    float mu  = s * (1.f / D);
    float var = ss * (1.f / D) - mu * mu;
    float inv = rsqrtf(var + 1e-5f);
    float4 gv = *(const float4*)(g    + lane * 4);
    float4 bv = *(const float4*)(beta + lane * 4);
    float4 o;
    o.x = gv.x * (t0 - mu) * inv + bv.x;
    o.y = gv.y * (t1 - mu) * inv + bv.y;
    o.z = gv.z * (t2 - mu) * inv + bv.z;
    o.w = gv.w * (t3 - mu) * inv + bv.w;
    *(float4*)(out + (size_t)row * D + lane * 4) = o;
}

// ---------------------------------------------------------------------------
extern "C" void kernel_launch(void* const* d_in, const int* in_sizes, int n_in,
                              void* d_out, int out_size, void* d_ws, size_t ws_size,
                              hipStream_t stream)
{
    const float* node_states = (const float*)d_in[0];
    const int*   edges       = (const int*)  d_in[1];
    const float* w_qkv = (const float*)d_in[2];
    const float* b_qkv = (const float*)d_in[3];
    const float* w_out = (const float*)d_in[4];
    const float* b_out = (const float*)d_in[5];
    const float* w1  = (const float*)d_in[6];
    const float* b1  = (const float*)d_in[7];
    const float* w2  = (const float*)d_in[8];
    const float* b2  = (const float*)d_in[9];
    const float* g1  = (const float*)d_in[10];
    const float* be1 = (const float*)d_in[11];
    const float* g2  = (const float*)d_in[12];
    const float* be2 = (const float*)d_in[13];

    const int N = in_sizes[0] / D;
    const int M = in_sizes[1] / 2;

    // workspace layout (floats); hbuf (N*FFSZ) aliases qkv+ex, both dead by then
    float* ws = (float*)d_ws;
    size_t off = 0;
    float*    qkv   = ws + off; off += (size_t)N * QKV3;
    float*    ex    = ws + off; off += (size_t)M * NH;
    unsigned* smaxU = (unsigned*)(ws + off); off += (size_t)N * NH;
    float*    ssum  = ws + off; off += (size_t)N * NH;
    float*    agg   = ws + off; off += (size_t)N * D;
    float*    msg   = ws + off; off += (size_t)N * D;
    float*    xb    = ws + off; off += (size_t)N * D;
    float*    ffb   = ws + off; off += (size_t)N * D;
    float*    hbuf  = ws;   // reuse of [qkv | ex] region: N*384 + M*8 >= N*512

    float* out_new = (float*)d_out;
    float* out_old = out_new + (size_t)N * D;

    // init scatter buffers
    {
        int n_head = N * NH, n_agg = N * D;
        init_kernel<<<(n_agg + 255) / 256, 256, 0, stream>>>(smaxU, ssum, agg,
                                                             n_head, n_agg);
    }

    // QKV projection: (N,384)
    gemm_bias_wmma<<<dim3(N / 16, QKV3 / 64), 128, 0, stream>>>(
        node_states, w_qkv, b_qkv, qkv, D, QKV3, 0);

    // attention scores + scatter softmax + aggregation
    long teh = (long)M * NH;
    edge_scores_kernel<<<(unsigned)((teh + 255) / 256), 256, 0, stream>>>(
        qkv, edges, smaxU, ex, M);
    edge_exp_kernel<<<(unsigned)((teh + 255) / 256), 256, 0, stream>>>(
        edges, smaxU, ssum, ex, M);
    long tag = (long)M * 32;
    edge_agg_kernel<<<(unsigned)((tag + 255) / 256), 256, 0, stream>>>(
        qkv, edges, ex, ssum, agg, M);

    // output projection: messages = agg @ w_out^T + b_out
    gemm_bias_wmma<<<dim3(N / 16, D / 64), 128, 0, stream>>>(
        agg, w_out, b_out, msg, D, D, 0);

    // x = LN(node_states + messages)
    ln_residual_kernel<<<(N + 7) / 8, 256, 0, stream>>>(
        node_states, msg, g1, be1, xb, N);

    // FFN: h = relu(x @ w1^T + b1) ; ff = h @ w2^T + b2
    gemm_bias_wmma<<<dim3(N / 16, FFSZ / 64), 128, 0, stream>>>(
        xb, w1, b1, hbuf, D, FFSZ, 1);
    gemm_bias_wmma<<<dim3(N / 16, D / 64), 128, 0, stream>>>(
        hbuf, w2, b2, ffb, FFSZ, D, 0);

    // out = LN(x + ff)
    ln_residual_kernel<<<(N + 7) / 8, 256, 0, stream>>>(
        xb, ffb, g2, be2, out_new, N);

    // second tuple element: old node states
    hipMemcpyAsync(out_old, node_states, (size_t)N * D * sizeof(float),
                   hipMemcpyDeviceToDevice, stream);
}